// SenderGAT_38774964748931
// MI455X (gfx1250) — compile-verified
//
#include <hip/hip_runtime.h>
#include <hip/hip_bf16.h>

typedef __attribute__((ext_vector_type(2))) float v2f;
typedef __attribute__((ext_vector_type(8))) float v8f;

#define N_NODES   50000
#define N_EDGES   800000
#define E_TOTAL   850000   // + self loops
#define F_IN      128
#define HEADS     4
#define C_DIM     64
#define HC        256      // HEADS*C
#define HIDDEN    128
#define NUM_GRAPHS 1000
#define NEG_SLOPE 0.2f
#define ENC_NEG_INF 0x007FFFFFu

// ---- orderable-uint encoding for float atomicMax (handles negatives) ----
__device__ __forceinline__ unsigned enc_f32(float f) {
  unsigned u = __float_as_uint(f);
  return (u & 0x80000000u) ? ~u : (u | 0x80000000u);
}
__device__ __forceinline__ float dec_f32(unsigned u) {
  u = (u & 0x80000000u) ? (u & 0x7FFFFFFFu) : ~u;
  return __uint_as_float(u);
}

// ================= K1: h = x @ W  (fp32 WMMA, one 16x16 tile per wave) ====
__global__ __launch_bounds__(256) void k_gemm_h(const float* __restrict__ x,
                                                const float* __restrict__ W,
                                                float* __restrict__ h) {
  const int wave  = threadIdx.x >> 5;
  const int lane  = threadIdx.x & 31;
  const int lane16 = lane & 15;
  const int lhi    = lane >> 4;
  const int tile = blockIdx.x * 8 + wave;      // 3125*16 = 50000 tiles exact
  const int mt = tile >> 4;                    // 16 n-tiles per tile-row
  const int nt = tile & 15;
  const int m0 = mt * 16;
  const int n0 = nt * 16;
  const float* arow = x + (long)(m0 + lane16) * F_IN;  // A: lane16 = M row
  const float* bcol = W + n0 + lane16;                 // B: lane16 = N col
  v8f acc = {};
  #pragma unroll 4
  for (int k0 = 0; k0 < F_IN; k0 += 4) {
    const int ka = k0 + 2 * lhi;               // k = k0 + v + 2*(lane/16)
    v2f a, b;
    a.x = arow[ka];
    a.y = arow[ka + 1];
    b.x = bcol[(long)ka * HC];
    b.y = bcol[(long)(ka + 1) * HC];
    acc = __builtin_amdgcn_wmma_f32_16x16x4_f32(false, a, false, b,
                                                (short)0, acc, false, false);
  }
  // C/D: VGPR v -> M = v + 8*(lane/16), N = lane16
  float* crow = h + n0 + lane16;
  #pragma unroll
  for (int v = 0; v < 8; v++)
    crow[(long)(m0 + v + 8 * lhi) * HC] = acc[v];
}

// ============ K2: per-(node,head) attention logits ========================
__global__ __launch_bounds__(256) void k_att(const float* __restrict__ h,
                                             const float* __restrict__ att_src,
                                             const float* __restrict__ att_dst,
                                             float* __restrict__ a_src,
                                             float* __restrict__ a_dst) {
  const int tid = blockIdx.x * 256 + threadIdx.x;
  if (tid >= N_NODES * HEADS) return;
  const int node = tid >> 2;
  const int head = tid & 3;
  const float* hp = h + (long)node * HC + head * C_DIM;
  const float* as = att_src + head * C_DIM;
  const float* ad = att_dst + head * C_DIM;
  float s = 0.f, d = 0.f;
  #pragma unroll 8
  for (int c = 0; c < C_DIM; c++) { const float v = hp[c]; s += v * as[c]; d += v * ad[c]; }
  a_src[tid] = s;
  a_dst[tid] = d;
}

// ============ K3a: init accumulators ======================================
__global__ __launch_bounds__(256) void k_init(float* __restrict__ accum,
                                              unsigned* __restrict__ emax,
                                              float* __restrict__ esum) {
  const int tid = blockIdx.x * 256 + threadIdx.x;
  if (tid < N_NODES * HC) accum[tid] = 0.f;
  if (tid < N_NODES * HEADS) { emax[tid] = ENC_NEG_INF; esum[tid] = 0.f; }
}

// ============ K3b: edge logits + segment max ==============================
__global__ __launch_bounds__(256) void k_edge_logits(const int* __restrict__ ei,
                                                     const float* __restrict__ a_src,
                                                     const float* __restrict__ a_dst,
                                                     float* __restrict__ e_buf,
                                                     unsigned* __restrict__ emax) {
  const int tid = blockIdx.x * 256 + threadIdx.x;
  if (tid >= E_TOTAL * HEADS) return;
  const int edge = tid >> 2;
  const int head = tid & 3;
  int s, d;
  if (edge < N_EDGES) { s = ei[edge]; d = ei[N_EDGES + edge]; }
  else { s = edge - N_EDGES; d = s; }             // self loop
  float e = a_src[s * HEADS + head] + a_dst[d * HEADS + head];
  e = e > 0.f ? e : NEG_SLOPE * e;                // leaky_relu
  e_buf[tid] = e;
  atomicMax(&emax[d * HEADS + head], enc_f32(e));
}

// ============ K3c: exp(e - max) + segment sum =============================
__global__ __launch_bounds__(256) void k_edge_exp(const int* __restrict__ ei,
                                                  float* __restrict__ e_buf,
                                                  const unsigned* __restrict__ emax,
                                                  float* __restrict__ esum) {
  const int tid = blockIdx.x * 256 + threadIdx.x;
  if (tid >= E_TOTAL * HEADS) return;
  const int edge = tid >> 2;
  const int head = tid & 3;
  int d;
  if (edge < N_EDGES) d = ei[N_EDGES + edge];
  else d = edge - N_EDGES;
  const float ex = expf(e_buf[tid] - dec_f32(emax[d * HEADS + head]));
  e_buf[tid] = ex;
  atomicAdd(&esum[d * HEADS + head], ex);
}

// ============ K3d: weighted scatter, one wave32 per edge ==================
__global__ __launch_bounds__(256) void k_scatter(const int* __restrict__ ei,
                                                 const float* __restrict__ e_buf,
                                                 const float* __restrict__ esum,
                                                 const float* __restrict__ h,
                                                 float* __restrict__ accum) {
  const int wave = threadIdx.x >> 5;
  const int lane = threadIdx.x & 31;
  const int edge = blockIdx.x * 8 + wave;
  if (edge >= E_TOTAL) return;
  int s, d;
  if (edge < N_EDGES) { s = ei[edge]; d = ei[N_EDGES + edge]; }
  else { s = edge - N_EDGES; d = s; }
  const float* hs = h + (long)s * HC;
  float* od = accum + (long)d * HC;
  #pragma unroll
  for (int head = 0; head < HEADS; head++) {
    const float alpha = e_buf[(long)edge * HEADS + head]
                      / (esum[d * HEADS + head] + 1e-16f);
    #pragma unroll
    for (int cc = 0; cc < 2; cc++) {
      const int c = head * C_DIM + cc * 32 + lane;
      atomicAdd(&od[c], alpha * hs[c]);         // resolves in 192MB L2
    }
  }
}

// ============ K4: gather target nodes, mean heads, + bias =================
__global__ __launch_bounds__(256) void k_target(const float* __restrict__ accum,
                                                const float* __restrict__ bias,
                                                const int* __restrict__ ptr,
                                                const int* __restrict__ tgt,
                                                float* __restrict__ t_emb) {
  const int tid = blockIdx.x * 256 + threadIdx.x;
  if (tid >= NUM_GRAPHS * C_DIM) return;
  const int g = tid >> 6;
  const int c = tid & 63;
  const int node = tgt[g] + ptr[g];
  const float* ap = accum + (long)node * HC + c;
  t_emb[tid] = 0.25f * (ap[0] + ap[64] + ap[128] + ap[192]) + bias[c];
}

// ============ K5: out = t_emb @ fc_W + fc_b  (fp32 WMMA) ==================
__global__ __launch_bounds__(256) void k_fc(const float* __restrict__ t_emb,
                                            const float* __restrict__ fc_W,
                                            const float* __restrict__ fc_b,
                                            float* __restrict__ out) {
  const int wave   = threadIdx.x >> 5;
  const int lane   = threadIdx.x & 31;
  const int lane16 = lane & 15;
  const int lhi    = lane >> 4;
  const int tile = blockIdx.x * 8 + wave;      // 63*8 = 504 tiles exact
  const int mt = tile >> 3;                    // HIDDEN/16 = 8 n-tiles
  const int nt = tile & 7;
  const int m0 = mt * 16;
  const int n0 = nt * 16;
  int mrow = m0 + lane16;
  if (mrow >= NUM_GRAPHS) mrow = NUM_GRAPHS - 1;   // clamp loads: EXEC stays full
  const float* arow = t_emb + (long)mrow * C_DIM;
  const float* bcol = fc_W + n0 + lane16;
  v8f acc = {};
  #pragma unroll
  for (int k0 = 0; k0 < C_DIM; k0 += 4) {
    const int ka = k0 + 2 * lhi;
    v2f a, b;
    a.x = arow[ka];
    a.y = arow[ka + 1];
    b.x = bcol[(long)ka * HIDDEN];
    b.y = bcol[(long)(ka + 1) * HIDDEN];
    acc = __builtin_amdgcn_wmma_f32_16x16x4_f32(false, a, false, b,
                                                (short)0, acc, false, false);
  }
  const float bv = fc_b[n0 + lane16];
  #pragma unroll
  for (int v = 0; v < 8; v++) {
    const int m = m0 + v + 8 * lhi;
    if (m < NUM_GRAPHS) out[(long)m * HIDDEN + n0 + lane16] = acc[v] + bv;
  }
}

extern "C" void kernel_launch(void* const* d_in, const int* in_sizes, int n_in,
                              void* d_out, int out_size, void* d_ws, size_t ws_size,
                              hipStream_t stream) {
  const float* x       = (const float*)d_in[0];
  const float* W       = (const float*)d_in[1];
  const float* att_src = (const float*)d_in[2];
  const float* att_dst = (const float*)d_in[3];
  const float* bias    = (const float*)d_in[4];
  const float* fc_W    = (const float*)d_in[5];
  const float* fc_b    = (const float*)d_in[6];
  const int*   ei      = (const int*)d_in[7];
  const int*   ptr     = (const int*)d_in[8];
  const int*   tgt     = (const int*)d_in[9];
  float* out = (float*)d_out;

  // ---- workspace carve-up (bytes) ----
  char* ws = (char*)d_ws;
  size_t off = 0;
  auto take = [&](size_t bytes) { void* p = ws + off; off = (off + bytes + 255) & ~(size_t)255; return p; };
  float*    h_buf  = (float*)   take((size_t)N_NODES * HC * 4);        // 51.2 MB
  float*    a_src  = (float*)   take((size_t)N_NODES * HEADS * 4);
  float*    a_dst  = (float*)   take((size_t)N_NODES * HEADS * 4);
  float*    e_buf  = (float*)   take((size_t)E_TOTAL * HEADS * 4);     // 13.6 MB
  unsigned* emax   = (unsigned*)take((size_t)N_NODES * HEADS * 4);
  float*    esum   = (float*)   take((size_t)N_NODES * HEADS * 4);
  float*    accum  = (float*)   take((size_t)N_NODES * HC * 4);        // 51.2 MB
  float*    t_emb  = (float*)   take((size_t)NUM_GRAPHS * C_DIM * 4);
  (void)ws_size; (void)in_sizes; (void)n_in; (void)out_size;

  // K1: h = x @ W                 (50000 tiles / 8 waves = 6250 blocks)
  k_gemm_h<<<6250, 256, 0, stream>>>(x, W, h_buf);
  // K3a: init accum/emax/esum    (12.8M threads)
  k_init<<<(N_NODES * HC + 255) / 256, 256, 0, stream>>>(accum, emax, esum);
  // K2: attention logits
  k_att<<<(N_NODES * HEADS + 255) / 256, 256, 0, stream>>>(h_buf, att_src, att_dst, a_src, a_dst);
  // K3b: edge logits + segment max
  k_edge_logits<<<(E_TOTAL * HEADS + 255) / 256, 256, 0, stream>>>(ei, a_src, a_dst, e_buf, emax);
  // K3c: exp + segment sum
  k_edge_exp<<<(E_TOTAL * HEADS + 255) / 256, 256, 0, stream>>>(ei, e_buf, emax, esum);
  // K3d: weighted scatter        (one wave per edge)
  k_scatter<<<(E_TOTAL + 7) / 8, 256, 0, stream>>>(ei, e_buf, esum, h_buf, accum);
  // K4: target gather + head mean + bias
  k_target<<<(NUM_GRAPHS * C_DIM + 255) / 256, 256, 0, stream>>>(accum, bias, ptr, tgt, t_emb);
  // K5: fc head                  (504 tiles / 8 waves = 63 blocks)
  k_fc<<<63, 256, 0, stream>>>(t_emb, fc_W, fc_b, out);
}